// ISPParameterGenerator_23708219474113
// MI455X (gfx1250) — compile-verified
//
#include <hip/hip_runtime.h>
#include <stdint.h>

// ---------------------------------------------------------------------------
// Expert scatter: out[e, w, :] = (e == idx[w]) ? emb[w, :] : 0
//   emb: [num_windows, 64] f32,  idx: [num_windows] i32,
//   out: [num_experts, num_windows, 64] f32
// Pure bandwidth problem (~256MB NT writes + 32MB reads -> ~12us @ 23.3TB/s).
// CDNA5 paths used: TDM tensor_load_to_lds (bulk DMA of the embedding tile
// into LDS, tracked with TENSORcnt), ds_load_b128 LDS reads, and
// non-temporal global_store_b128 for the streaming output.
// ---------------------------------------------------------------------------

typedef float        f32x4 __attribute__((ext_vector_type(4)));
typedef unsigned int u32x4 __attribute__((ext_vector_type(4)));
typedef int          i32x8 __attribute__((ext_vector_type(8)));
typedef int          i32x4 __attribute__((ext_vector_type(4)));

#define DIM            64
#define WIN_PER_BLOCK  64
#define NTHREADS       256

__global__ __launch_bounds__(NTHREADS)
void isp_expert_scatter_kernel(const float* __restrict__ src,
                               const int*   __restrict__ eidx,
                               float*       __restrict__ out,
                               int num_windows, int num_experts)
{
    // One 64x64 f32 tile (16KB), vector-aligned for ds_load_b128.
    __shared__ f32x4 tile[WIN_PER_BLOCK * DIM / 4];

    const int tid = threadIdx.x;
    const int w0  = blockIdx.x * WIN_PER_BLOCK;
    int rows = num_windows - w0;
    if (rows > WIN_PER_BLOCK) rows = WIN_PER_BLOCK;

    // ---- Wave 0 issues the TDM bulk copy: rows x 64 f32, global -> LDS ----
    if (tid < 32) {
        uint64_t gaddr   = (uint64_t)(uintptr_t)(src + (size_t)w0 * DIM);
        uint32_t lds_off = (uint32_t)(uintptr_t)(&tile[0]); // low 32 bits of flat = LDS offset

        // D# group 0: count=1 | lds_addr | global_addr[56:0] | type=2
        u32x4 g0;
        g0.x = 1u;
        g0.y = lds_off;
        g0.z = (uint32_t)gaddr;
        g0.w = (uint32_t)((gaddr >> 32) & 0x01FFFFFFu) | (2u << 30);

        // D# group 1 (bitfields per CDNA5 ISA):
        //   w0: workgroup_mask=0, data_size=2 (4B), no barrier/iterate/pad
        //   w1[31:16] = tensor_dim0 = 64
        //   w2[31:16] = tensor_dim1 = rows
        //   w3[31:16] = tile_dim0   = 64
        //   w4[15:0]  = tile_dim1   = rows (tile_dim2 = 0)
        //   w5        = tensor_dim0_stride = 64
        i32x8 g1;
        g1[0] = (2 << 16);
        g1[1] = (DIM << 16);
        g1[2] = (rows << 16);
        g1[3] = (DIM << 16);
        g1[4] = rows;
        g1[5] = DIM;
        g1[6] = 0;
        g1[7] = 0;

        i32x4 gz4 = {0, 0, 0, 0};
        i32x8 gz8 = {0, 0, 0, 0, 0, 0, 0, 0};
        // 6-arg form (clang-23 / therock-10.0 toolchain):
        //   (g0, g1, g2, g3, extra_group, cpol)
        __builtin_amdgcn_tensor_load_to_lds(g0, g1, gz4, gz4, gz8, 0);
        // TENSORcnt is per-wave: drain before publishing via the barrier.
        __builtin_amdgcn_s_wait_tensorcnt(0);
    }
    __syncthreads();

    // ---- Scatter: each thread owns float4 chunks; 16 lanes cover one row ----
    const f32x4 zero = {0.f, 0.f, 0.f, 0.f};
    const size_t estride = (size_t)num_windows * DIM;

    #pragma unroll
    for (int r = 0; r < (WIN_PER_BLOCK * DIM / 4) / NTHREADS; ++r) {
        const int lin = r * NTHREADS + tid;   // 0..1023
        const int wl  = lin >> 4;             // local window
        const int c   = lin & 15;             // float4 chunk within row
        const int w   = w0 + wl;
        if (w < num_windows) {
            const int   e = eidx[w];          // 16 lanes share one cached load
            const f32x4 v = tile[lin];        // ds_load_b128, conflict-free
            float* base = out + (size_t)w * DIM + (size_t)c * 4;
            for (int ee = 0; ee < num_experts; ++ee) {
                f32x4 o = (ee == e) ? v : zero;
                __builtin_nontemporal_store(o, (f32x4*)(base + (size_t)ee * estride));
            }
        }
    }
}

extern "C" void kernel_launch(void* const* d_in, const int* in_sizes, int n_in,
                              void* d_out, int out_size, void* d_ws, size_t ws_size,
                              hipStream_t stream)
{
    const float* src  = (const float*)d_in[0];   // isp_per_win  [B,W,K,64] f32
    const int*   eidx = (const int*)d_in[1];     // expert_indices [B*W*K] i32
    float*       out  = (float*)d_out;           // [E, B*W*K, 64] f32

    const int num_windows = in_sizes[1];                       // B*W*K (K=1)
    const int num_experts = out_size / (num_windows * DIM);    // = 8

    const int blocks = (num_windows + WIN_PER_BLOCK - 1) / WIN_PER_BLOCK;
    isp_expert_scatter_kernel<<<blocks, NTHREADS, 0, stream>>>(
        src, eidx, out, num_windows, num_experts);
}